// GCN_29643864277073
// MI455X (gfx1250) — compile-verified
//
#include <hip/hip_runtime.h>

typedef float v2f __attribute__((ext_vector_type(2)));
typedef float v8f __attribute__((ext_vector_type(8)));

#define DIN   256
#define HID   64
#define NCLS  47
#define NCLSP 48   // padded layer-2 width (3 WMMA column tiles)

// ---------------- utility ----------------
__global__ void fill0_kernel(float* __restrict__ p, long long n) {
  long long i = (long long)blockIdx.x * blockDim.x + threadIdx.x;
  if (i < n) p[i] = 0.0f;
}

// deg[dst] += 1 per edge
__global__ void degree_kernel(const long long* __restrict__ dst,
                              float* __restrict__ deg, long long E) {
  long long e = (long long)blockIdx.x * blockDim.x + threadIdx.x;
  if (e < E) atomicAdd(deg + dst[e], 1.0f);
}

// dinv = rsqrt(deg + 1)  (self loop; deg >= 1 always afterward)
__global__ void dinv_kernel(float* __restrict__ deg, long long n) {
  long long i = (long long)blockIdx.x * blockDim.x + threadIdx.x;
  if (i < n) deg[i] = rsqrtf(deg[i] + 1.0f);
}

// ---------------- Layer 1 GEMM: h1[n,64] = x[n,256] @ W1[256,64] ----------------
// One wave per 16-row stripe; 4 column tiles (64 cols) accumulated in-register.
__global__ void gemm1_wmma(const float* __restrict__ x, const float* __restrict__ w1,
                           float* __restrict__ h1, long long n) {
  const int lane   = threadIdx.x;          // blockDim.x == 32
  const long long m_base = (long long)blockIdx.x * 16;
  const int ml   = lane & 15;
  const int ksub = (lane >> 4) << 1;       // 0 or 2
  long long arowi = m_base + ml;
  if (arowi >= n) arowi = n - 1;           // clamp (n % 16 == 0 in practice)
  const float* arow = x + arowi * DIN + ksub;

  v8f c0 = {}, c1 = {}, c2 = {}, c3 = {};
  for (int k0 = 0; k0 < DIN; k0 += 4) {
    v2f a = *(const v2f*)(arow + k0);      // A: K = k0+ksub, k0+ksub+1
    const int kb = k0 + ksub;
    const float* w0 = w1 + (long long)kb * HID;
    const float* w1r = w0 + HID;
    v2f b0, b1, b2, b3;
    b0.x = w0[ 0 + ml]; b0.y = w1r[ 0 + ml];
    b1.x = w0[16 + ml]; b1.y = w1r[16 + ml];
    b2.x = w0[32 + ml]; b2.y = w1r[32 + ml];
    b3.x = w0[48 + ml]; b3.y = w1r[48 + ml];
    c0 = __builtin_amdgcn_wmma_f32_16x16x4_f32(false, a, false, b0, (short)0, c0, false, false);
    c1 = __builtin_amdgcn_wmma_f32_16x16x4_f32(false, a, false, b1, (short)0, c1, false, false);
    c2 = __builtin_amdgcn_wmma_f32_16x16x4_f32(false, a, false, b2, (short)0, c2, false, false);
    c3 = __builtin_amdgcn_wmma_f32_16x16x4_f32(false, a, false, b3, (short)0, c3, false, false);
  }
  // C/D layout: VGPR r -> M = r (lanes 0-15), M = r+8 (lanes 16-31); N = lane&15
  const int m_off = (lane >> 4) * 8;
  #pragma unroll
  for (int r = 0; r < 8; ++r) {
    long long row = m_base + m_off + r;
    if (row < n) {
      float* hrow = h1 + row * HID;
      hrow[ 0 + ml] = c0[r];
      hrow[16 + ml] = c1[r];
      hrow[32 + ml] = c2[r];
      hrow[48 + ml] = c3[r];
    }
  }
}

// ---------------- Layer 1 edge scatter: agg1[dst] += h1[src] * norm ----------------
__global__ void scatter1_kernel(const long long* __restrict__ src,
                                const long long* __restrict__ dst,
                                const float* __restrict__ dinv,
                                const float* __restrict__ h1,
                                float* __restrict__ agg1, long long E) {
  long long idx = (long long)blockIdx.x * blockDim.x + threadIdx.x;
  long long e = idx >> 6;                  // HID = 64 features per edge
  if (e >= E) return;
  const int f = (int)(idx & 63);
  const long long s = src[e], d = dst[e];
  const float nm = dinv[s] * dinv[d];
  atomicAdd(agg1 + d * HID + f, h1[s * HID + f] * nm);
}

// self-loop + bias + ReLU:  agg1 = relu(agg1 + h1*dinv^2 + b1)
__global__ void post1_kernel(float* __restrict__ agg1, const float* __restrict__ h1,
                             const float* __restrict__ dinv, const float* __restrict__ b1,
                             long long total) {
  long long idx = (long long)blockIdx.x * blockDim.x + threadIdx.x;
  if (idx >= total) return;
  const long long i = idx >> 6;
  const int f = (int)(idx & 63);
  const float di = dinv[i];
  float v = agg1[idx] + h1[idx] * di * di + b1[f];
  agg1[idx] = v > 0.0f ? v : 0.0f;
}

// ---------------- Layer 2 GEMM: h2[n,48] = agg1[n,64] @ W2[64,47] (col 47 zero-padded) ----
__global__ void gemm2_wmma(const float* __restrict__ ag, const float* __restrict__ w2,
                           float* __restrict__ h2, long long n) {
  const int lane   = threadIdx.x;          // blockDim.x == 32
  const long long m_base = (long long)blockIdx.x * 16;
  const int ml   = lane & 15;
  const int ksub = (lane >> 4) << 1;
  long long arowi = m_base + ml;
  if (arowi >= n) arowi = n - 1;
  const float* arow = ag + arowi * HID + ksub;
  const bool lastcol_ok = (32 + ml) < NCLS;  // guard padded column 47

  v8f c0 = {}, c1 = {}, c2 = {};
  for (int k0 = 0; k0 < HID; k0 += 4) {
    v2f a = *(const v2f*)(arow + k0);
    const int kb = k0 + ksub;
    const float* w0 = w2 + (long long)kb * NCLS;
    const float* w1r = w0 + NCLS;
    v2f b0, b1, b2;
    b0.x = w0[ 0 + ml]; b0.y = w1r[ 0 + ml];
    b1.x = w0[16 + ml]; b1.y = w1r[16 + ml];
    b2.x = lastcol_ok ? w0[32 + ml]  : 0.0f;
    b2.y = lastcol_ok ? w1r[32 + ml] : 0.0f;
    c0 = __builtin_amdgcn_wmma_f32_16x16x4_f32(false, a, false, b0, (short)0, c0, false, false);
    c1 = __builtin_amdgcn_wmma_f32_16x16x4_f32(false, a, false, b1, (short)0, c1, false, false);
    c2 = __builtin_amdgcn_wmma_f32_16x16x4_f32(false, a, false, b2, (short)0, c2, false, false);
  }
  const int m_off = (lane >> 4) * 8;
  #pragma unroll
  for (int r = 0; r < 8; ++r) {
    long long row = m_base + m_off + r;
    if (row < n) {
      float* hrow = h2 + row * NCLSP;
      hrow[ 0 + ml] = c0[r];
      hrow[16 + ml] = c1[r];
      hrow[32 + ml] = c2[r];
    }
  }
}

// ---------------- Layer 2 edge scatter: out[dst] += h2[src] * norm ----------------
__global__ void scatter2_kernel(const long long* __restrict__ src,
                                const long long* __restrict__ dst,
                                const float* __restrict__ dinv,
                                const float* __restrict__ h2,
                                float* __restrict__ out, long long E) {
  long long idx = (long long)blockIdx.x * blockDim.x + threadIdx.x;
  long long e = idx / NCLSP;
  if (e >= E) return;
  const int f = (int)(idx - e * NCLSP);
  if (f >= NCLS) return;
  const long long s = src[e], d = dst[e];
  const float nm = dinv[s] * dinv[d];
  atomicAdd(out + d * NCLS + f, h2[s * NCLSP + f] * nm);
}

// self-loop + bias:  out += h2*dinv^2 + b2
__global__ void post2_kernel(float* __restrict__ out, const float* __restrict__ h2,
                             const float* __restrict__ dinv, const float* __restrict__ b2,
                             long long total) {
  long long idx = (long long)blockIdx.x * blockDim.x + threadIdx.x;
  if (idx >= total) return;
  const long long i = idx / NCLS;
  const int f = (int)(idx - i * NCLS);
  const float di = dinv[i];
  out[idx] += h2[i * NCLSP + f] * di * di + b2[f];
}

extern "C" void kernel_launch(void* const* d_in, const int* in_sizes, int n_in,
                              void* d_out, int out_size, void* d_ws, size_t ws_size,
                              hipStream_t stream) {
  const float*     x    = (const float*)d_in[0];
  const long long* eidx = (const long long*)d_in[1];   // int64 per reference
  const float*     W1   = (const float*)d_in[2];
  const float*     b1   = (const float*)d_in[3];
  const float*     W2   = (const float*)d_in[4];
  const float*     b2   = (const float*)d_in[5];
  float* out = (float*)d_out;

  const long long n = (long long)in_sizes[0] / DIN;    // 100000
  const long long E = (long long)in_sizes[1] / 2;      // 1600000
  const long long* src = eidx;
  const long long* dst = eidx + E;

  // workspace: dinv[n] | h1[n*64] | agg1[n*64] | h2[n*48]  (~71 MB)
  float* dinv = (float*)d_ws;
  float* h1   = dinv + n;
  float* agg1 = h1 + n * HID;
  float* h2   = agg1 + n * HID;

  const int B = 256;
  const long long nb  = (n + B - 1) / B;
  const long long mb  = (n + 15) / 16;

  // normalization
  fill0_kernel <<<(unsigned)nb, B, 0, stream>>>(dinv, n);
  degree_kernel<<<(unsigned)((E + B - 1) / B), B, 0, stream>>>(dst, dinv, E);
  dinv_kernel  <<<(unsigned)nb, B, 0, stream>>>(dinv, n);

  // layer 1
  gemm1_wmma   <<<(unsigned)mb, 32, 0, stream>>>(x, W1, h1, n);
  fill0_kernel <<<(unsigned)((n * HID + B - 1) / B), B, 0, stream>>>(agg1, n * HID);
  scatter1_kernel<<<(unsigned)((E * HID + B - 1) / B), B, 0, stream>>>(src, dst, dinv, h1, agg1, E);
  post1_kernel <<<(unsigned)((n * HID + B - 1) / B), B, 0, stream>>>(agg1, h1, dinv, b1, n * HID);

  // layer 2
  gemm2_wmma   <<<(unsigned)mb, 32, 0, stream>>>(agg1, W2, h2, n);
  fill0_kernel <<<(unsigned)(((long long)out_size + B - 1) / B), B, 0, stream>>>(out, out_size);
  scatter2_kernel<<<(unsigned)((E * NCLSP + B - 1) / B), B, 0, stream>>>(src, dst, dinv, h2, out, E);
  post2_kernel <<<(unsigned)((n * NCLS + B - 1) / B), B, 0, stream>>>(out, h2, dinv, b2, n * NCLS);
}